// GraphSuperResolutionNet_70781061038080
// MI455X (gfx1250) — compile-verified
//
#include <hip/hip_runtime.h>
#include <hip/hip_bf16.h>
#include <math.h>

// Problem dims (fixed by the reference)
#define B_   32
#define H_   256
#define W_   256
#define HW_  (H_*W_)
#define NP_  (B_*HW_)          // 2,097,152 pixels
#define KP_  8
#define HS_  (H_/KP_)          // 32
#define WS_  (W_/KP_)          // 32
#define CG_ITERS 500

typedef __attribute__((ext_vector_type(2))) float v2f;
typedef __attribute__((ext_vector_type(8))) float v8f;

// ---------------------------------------------------------------------------
// CDNA5 helpers: async global->LDS copy + wait (gfx1250 path, guarded)
// ---------------------------------------------------------------------------
__device__ __forceinline__ void async_copy_f32(const float* gsrc, float* ldst) {
#if defined(__gfx1250__) && __has_builtin(__builtin_amdgcn_global_load_async_to_lds_b32)
  __builtin_amdgcn_global_load_async_to_lds_b32(
      (__attribute__((address_space(1))) int*)gsrc,
      (__attribute__((address_space(3))) int*)ldst, 0, 0);
#else
  *ldst = *gsrc;
#endif
}

__device__ __forceinline__ void wait_async_all() {
#if defined(__gfx1250__) && __has_builtin(__builtin_amdgcn_s_wait_asynccnt)
  __builtin_amdgcn_s_wait_asynccnt(0);
#endif
}

// WMMA f32 16x16x4 (FP32 matrix core path; required precision for CG)
__device__ __forceinline__ v8f wmma_f32_16x16x4(v2f a, v2f b, v8f c) {
  return __builtin_amdgcn_wmma_f32_16x16x4_f32(
      /*neg_a=*/false, a, /*neg_b=*/false, b,
      /*c_mod=*/(short)0, c, /*reuse_a=*/false, /*reuse_b=*/false);
}

// 256-thread block reduction (wave32). Result valid on thread 0.
__device__ __forceinline__ float block_sum256(float v) {
  __shared__ float red[8];
  for (int o = 16; o > 0; o >>= 1) v += __shfl_down(v, o, 32);
  int lane = threadIdx.x & 31, wid = threadIdx.x >> 5;
  if (lane == 0) red[wid] = v;
  __syncthreads();
  float s = 0.f;
  if (wid == 0) {
    s = (lane < 8) ? red[lane] : 0.f;
    for (int o = 4; o > 0; o >>= 1) s += __shfl_down(s, o, 32);
  }
  return s;
}

__device__ __forceinline__ float pw15(float x) {  // |x|^1.5
  float a = fabsf(x);
  return a * sqrtf(a);
}

// ---------------------------------------------------------------------------
// 1. Neighbor affinity  (writes aff[B][5][H][W], already scaled by lambda)
// ---------------------------------------------------------------------------
__global__ void aff_kernel(const float* __restrict__ guide,
                           const float* __restrict__ log_lam,
                           const float* __restrict__ log_mu,
                           float* __restrict__ aff) {
  int i = blockIdx.x * blockDim.x + threadIdx.x;
  int w = i & (W_ - 1);
  int h = (i >> 8) & (H_ - 1);
  int b = i >> 16;
  float lam = __expf(log_lam[0]);
  float mu  = __expf(log_mu[0]);
  float inv = 1.0f / (1e-6f + mu * mu);
  const float* g = guide + (size_t)b * 3 * HW_ + (size_t)h * W_ + w;
  float c0 = g[0], c1 = g[HW_], c2 = g[2 * HW_];

  float d0, d1, d2, d3;
  { float n0 = 0, n1 = 0, n2 = 0;                      // top
    if (h > 0) { n0 = g[-W_]; n1 = g[HW_ - W_]; n2 = g[2 * HW_ - W_]; }
    d0 = (pw15(n0 - c0) + pw15(n1 - c1) + pw15(n2 - c2)) * (1.f / 3.f); }
  { float n0 = 0, n1 = 0, n2 = 0;                      // bottom
    if (h < H_ - 1) { n0 = g[W_]; n1 = g[HW_ + W_]; n2 = g[2 * HW_ + W_]; }
    d1 = (pw15(n0 - c0) + pw15(n1 - c1) + pw15(n2 - c2)) * (1.f / 3.f); }
  { float n0 = 0, n1 = 0, n2 = 0;                      // left
    if (w > 0) { n0 = g[-1]; n1 = g[HW_ - 1]; n2 = g[2 * HW_ - 1]; }
    d2 = (pw15(n0 - c0) + pw15(n1 - c1) + pw15(n2 - c2)) * (1.f / 3.f); }
  { float n0 = 0, n1 = 0, n2 = 0;                      // right
    if (w < W_ - 1) { n0 = g[1]; n1 = g[HW_ + 1]; n2 = g[2 * HW_ + 1]; }
    d3 = (pw15(n0 - c0) + pw15(n1 - c1) + pw15(n2 - c2)) * (1.f / 3.f); }

  float a0 = __expf(-d0 * inv) * ((h > 0) ? 1.f : 0.f);
  float a1 = __expf(-d1 * inv) * ((h < H_ - 1) ? 1.f : 0.f);
  float a2 = __expf(-d2 * inv) * ((w > 0) ? 1.f : 0.f);
  float a3 = __expf(-d3 * inv) * ((w < W_ - 1) ? 1.f : 0.f);
  float ctr = a0 + a1 + a2 + a3;

  float* o = aff + (size_t)b * 5 * HW_ + (size_t)h * W_ + w;
  o[0]        = lam * a0;
  o[HW_]      = lam * a1;
  o[2 * HW_]  = lam * a2;
  o[3 * HW_]  = lam * a3;
  o[4 * HW_]  = lam * ctr;
}

// ---------------------------------------------------------------------------
// 2. t = L y  (variable-weight 5-point stencil). 32x32 tile + halo in LDS,
//    staged with async global->LDS loads. blockDim = 256 (32x8, 4 rows/thr)
// ---------------------------------------------------------------------------
#define TS_ 32
__global__ void stencil_kernel(const float* __restrict__ y,
                               const float* __restrict__ aff,
                               float* __restrict__ t) {
  __shared__ float sm[(TS_ + 2) * (TS_ + 2)];
  int tile = blockIdx.x;                  // b*64 + ty*8 + tx
  int b  = tile >> 6;
  int ty = (tile >> 3) & 7, tx = tile & 7;
  int y0 = ty * TS_, x0 = tx * TS_;
  const float* yb = y + (size_t)b * HW_;

  for (int i = threadIdx.x; i < (TS_ + 2) * (TS_ + 2); i += blockDim.x) {
    int ly = i / (TS_ + 2), lx = i - ly * (TS_ + 2);
    int gy = y0 + ly - 1, gx = x0 + lx - 1;
    if ((unsigned)gy < (unsigned)H_ && (unsigned)gx < (unsigned)W_)
      async_copy_f32(yb + (size_t)gy * W_ + gx, &sm[i]);
    else
      sm[i] = 0.f;
  }
  wait_async_all();
  __syncthreads();

  int lx  = threadIdx.x & 31;
  int lyb = threadIdx.x >> 5;             // 0..7
  for (int rr = 0; rr < 4; ++rr) {
    int ly = lyb * 4 + rr;                // 0..31
    int gy = y0 + ly, gx = x0 + lx;
    size_t pix = (size_t)b * HW_ + (size_t)gy * W_ + gx;
    const float* a = aff + (size_t)b * 5 * HW_ + (size_t)gy * W_ + gx;
    float c   = sm[(ly + 1) * (TS_ + 2) + (lx + 1)];
    float top = sm[(ly    ) * (TS_ + 2) + (lx + 1)];
    float bot = sm[(ly + 2) * (TS_ + 2) + (lx + 1)];
    float lef = sm[(ly + 1) * (TS_ + 2) + (lx    )];
    float rig = sm[(ly + 1) * (TS_ + 2) + (lx + 2)];
    float wt = a[0], wb = a[HW_], wl = a[2 * HW_], wr = a[3 * HW_], wc = a[4 * HW_];
    t[pix] = wc * c - wt * top - wb * bot - wl * lef - wr * rig;
  }
}

// ---------------------------------------------------------------------------
// 3. WMMA pooling stage 1:  Trow[b][r][g] = sum_{j in 8g..8g+7} t[b][r][j]
//    As GEMM: Trow = t(256x256) x O(256x32), K-chunked by 4 (chunk stays in
//    one pool group), f32 WMMA 16x16x4. One wave per 16x16 output tile.
// ---------------------------------------------------------------------------
__global__ void pool1_kernel(const float* __restrict__ t,
                             float* __restrict__ Trow) {
  int wv   = blockIdx.x * (blockDim.x >> 5) + (threadIdx.x >> 5); // 0..B*32-1
  int b    = wv >> 5;
  int tIdx = wv & 31;                     // 16 M-tiles x 2 N-tiles
  int mt   = tIdx >> 1, nt = tIdx & 1;
  int lane = threadIdx.x & 31;
  int m    = mt * 16 + (lane & 15);
  int n    = nt * 16 + (lane & 15);
  int ksub = (lane >> 4) * 2;             // A layout: K=0,1 / K=2,3 split
  const float* tb = t + (size_t)b * HW_ + (size_t)m * W_;

  v8f c = {0.f, 0.f, 0.f, 0.f, 0.f, 0.f, 0.f, 0.f};
  for (int k0 = 0; k0 < W_; k0 += 4) {
    v2f a; a.x = tb[k0 + ksub]; a.y = tb[k0 + ksub + 1];
    float bv = (n == (k0 >> 3)) ? 1.f : 0.f;   // operator O tile (splat: rows equal)
    v2f bb; bb.x = bv; bb.y = bv;
    c = wmma_f32_16x16x4(a, bb, c);
  }
  float* Tb = Trow + (size_t)b * H_ * WS_;
  int mrow = mt * 16 + ((lane >> 4) * 8);      // C layout: M=v / M=v+8 halves
#pragma unroll
  for (int v = 0; v < 8; ++v)
    Tb[(size_t)(mrow + v) * WS_ + n] = c[v];
}

// ---------------------------------------------------------------------------
// 4. WMMA pooling stage 2:  S = (1/64) * O^T(32x256) x Trow(256x32)
// ---------------------------------------------------------------------------
__global__ void pool2_kernel(const float* __restrict__ Trow,
                             float* __restrict__ S) {
  int wv   = blockIdx.x * (blockDim.x >> 5) + (threadIdx.x >> 5); // 0..B*4-1
  int b    = wv >> 2;
  int tIdx = wv & 3;
  int mt   = tIdx >> 1, nt = tIdx & 1;
  int lane = threadIdx.x & 31;
  int mG   = mt * 16 + (lane & 15);
  int n    = nt * 16 + (lane & 15);
  int rb   = (lane >> 4) * 2;             // B layout: rows {0,1} / {2,3} split
  const float* Tb = Trow + (size_t)b * H_ * WS_;

  v8f c = {0.f, 0.f, 0.f, 0.f, 0.f, 0.f, 0.f, 0.f};
  for (int k0 = 0; k0 < H_; k0 += 4) {
    float av = ((k0 >> 3) == mG) ? 1.f : 0.f;  // operator O^T tile
    v2f a; a.x = av; a.y = av;
    v2f bb;
    bb.x = Tb[(size_t)(k0 + rb)     * WS_ + n];
    bb.y = Tb[(size_t)(k0 + rb + 1) * WS_ + n];
    c = wmma_f32_16x16x4(a, bb, c);
  }
  float* Sb = S + (size_t)b * HS_ * WS_;
  int mrow = mt * 16 + ((lane >> 4) * 8);
#pragma unroll
  for (int v = 0; v < 8; ++v)
    Sb[(size_t)(mrow + v) * WS_ + n] = c[v] * (1.0f / 64.0f);
}

// ---------------------------------------------------------------------------
// Elementwise CG kernels, float4-vectorized (4 consecutive pixels share the
// same 8x8 pooling block, so one S load per float4). Grid: NP_/4/256 blocks.
// ---------------------------------------------------------------------------

// 5. y init:  y = upsample(source)
__global__ void upsample_kernel(const float* __restrict__ src,
                                float* __restrict__ y) {
  int i = blockIdx.x * blockDim.x + threadIdx.x;      // float4 index
  int base = i << 2;
  int w = base & (W_ - 1);
  int h = (base >> 8) & (H_ - 1);
  int b = base >> 16;
  float s = src[(size_t)b * HS_ * WS_ + (h >> 3) * WS_ + (w >> 3)];
  ((float4*)y)[i] = make_float4(s, s, s, s);
}

// 6. r = p = up(S) - t   (b-vector of the reduced system); rr += r.r
__global__ void init_resid_kernel(const float* __restrict__ t,
                                  const float* __restrict__ S,
                                  float* __restrict__ r,
                                  float* __restrict__ p,
                                  float* __restrict__ sc) {
  int i = blockIdx.x * blockDim.x + threadIdx.x;
  int base = i << 2;
  int w = base & (W_ - 1);
  int h = (base >> 8) & (H_ - 1);
  int b = base >> 16;
  float s = S[(size_t)b * HS_ * WS_ + (h >> 3) * WS_ + (w >> 3)];
  float4 tv = ((const float4*)t)[i];
  float4 rv = make_float4(s - tv.x, s - tv.y, s - tv.z, s - tv.w);
  ((float4*)r)[i] = rv;
  ((float4*)p)[i] = rv;
  float sum = block_sum256(rv.x * rv.x + rv.y * rv.y + rv.z * rv.z + rv.w * rv.w);
  if (threadIdx.x == 0) atomicAdd(&sc[0], sum);
}

// 7. Ap = t - up(S) (projection), overwrite t; pAp += p.Ap
__global__ void finish_kernel(const float* __restrict__ p,
                              float* __restrict__ t,
                              const float* __restrict__ S,
                              float* __restrict__ sc) {
  int i = blockIdx.x * blockDim.x + threadIdx.x;
  int base = i << 2;
  int w = base & (W_ - 1);
  int h = (base >> 8) & (H_ - 1);
  int b = base >> 16;
  float s = S[(size_t)b * HS_ * WS_ + (h >> 3) * WS_ + (w >> 3)];
  float4 tv = ((const float4*)t)[i];
  float4 pv = ((const float4*)p)[i];
  float4 ap = make_float4(tv.x - s, tv.y - s, tv.z - s, tv.w - s);
  ((float4*)t)[i] = ap;
  float sum = block_sum256(pv.x * ap.x + pv.y * ap.y + pv.z * ap.z + pv.w * ap.w);
  if (threadIdx.x == 0) atomicAdd(&sc[1], sum);
}

// 8. scalar stage 1: alpha = rr/pAp ; zero rr_new
__global__ void scalar1_kernel(float* sc) {
  float rr = sc[0], pAp = sc[1];
  sc[2] = (fabsf(pAp) > 1e-30f) ? rr / pAp : 0.f;   // alpha
  sc[4] = 0.f;                                      // rr_new accumulator
}

// 9. y += alpha p ; r -= alpha Ap ; rr_new += r.r
__global__ void axpy_kernel(float* __restrict__ y,
                            float* __restrict__ r,
                            const float* __restrict__ p,
                            const float* __restrict__ Ap,
                            const float* __restrict__ sc,
                            float* __restrict__ sc_rrnew) {
  int i = blockIdx.x * blockDim.x + threadIdx.x;
  float alpha = sc[2];
  float4 yv = ((const float4*)y)[i];
  float4 pv = ((const float4*)p)[i];
  float4 av = ((const float4*)Ap)[i];
  float4 rv = ((const float4*)r)[i];
  yv.x += alpha * pv.x; yv.y += alpha * pv.y;
  yv.z += alpha * pv.z; yv.w += alpha * pv.w;
  rv.x -= alpha * av.x; rv.y -= alpha * av.y;
  rv.z -= alpha * av.z; rv.w -= alpha * av.w;
  ((float4*)y)[i] = yv;
  ((float4*)r)[i] = rv;
  float sum = block_sum256(rv.x * rv.x + rv.y * rv.y + rv.z * rv.z + rv.w * rv.w);
  if (threadIdx.x == 0) atomicAdd(sc_rrnew, sum);
}

// 10. scalar stage 2: beta = rr_new/rr ; rr = rr_new ; zero pAp
__global__ void scalar2_kernel(float* sc) {
  float rr = sc[0], rrn = sc[4];
  sc[3] = (fabsf(rr) > 1e-30f) ? rrn / rr : 0.f;    // beta
  sc[0] = rrn;
  sc[1] = 0.f;
}

// 11. p = r + beta p
__global__ void pupdate_kernel(float* __restrict__ p,
                               const float* __restrict__ r,
                               const float* __restrict__ sc) {
  int i = blockIdx.x * blockDim.x + threadIdx.x;
  float beta = sc[3];
  float4 rv = ((const float4*)r)[i];
  float4 pv = ((const float4*)p)[i];
  pv.x = rv.x + beta * pv.x; pv.y = rv.y + beta * pv.y;
  pv.z = rv.z + beta * pv.z; pv.w = rv.w + beta * pv.w;
  ((float4*)p)[i] = pv;
}

__global__ void init_scalars_kernel(float* sc) {
  if (threadIdx.x < 8) sc[threadIdx.x] = 0.f;
}

// ---------------------------------------------------------------------------
extern "C" void kernel_launch(void* const* d_in, const int* in_sizes, int n_in,
                              void* d_out, int out_size, void* d_ws, size_t ws_size,
                              hipStream_t stream) {
  (void)in_sizes; (void)n_in; (void)out_size; (void)ws_size;
  const float* guide  = (const float*)d_in[0];
  const float* source = (const float*)d_in[1];
  const float* logl   = (const float*)d_in[2];
  const float* logm   = (const float*)d_in[3];

  float* yout = (float*)d_out;            // [B][1][H][W]
  float* aff  = yout + (size_t)NP_;       // [B][5][H][W]

  float* ws   = (float*)d_ws;
  float* p    = ws;
  float* r    = ws + (size_t)NP_;
  float* t    = ws + 2 * (size_t)NP_;                 // t, then reused as Ap
  float* Trow = ws + 3 * (size_t)NP_;                 // [B][256][32]
  float* S    = Trow + (size_t)B_ * H_ * WS_;         // [B][32][32]
  float* sc   = S + (size_t)B_ * HS_ * WS_;           // rr,pAp,alpha,beta,rr_new

  const int EB  = 256;             // elementwise block
  const int EG  = NP_ / EB;        // scalar-grid (aff): 8192 blocks
  const int EG4 = NP_ / (EB * 4);  // float4-grid: 2048 blocks
  const int SG  = B_ * 64;         // stencil tiles: 2048

  init_scalars_kernel<<<1, 32, 0, stream>>>(sc);
  aff_kernel<<<EG, EB, 0, stream>>>(guide, logl, logm, aff);
  upsample_kernel<<<EG4, EB, 0, stream>>>(source, yout);    // y = y_p

  // b-vector: r = p = up(down(L y_p)) - L y_p
  stencil_kernel<<<SG, 256, 0, stream>>>(yout, aff, t);
  pool1_kernel<<<(B_ * 32) / 4, 128, 0, stream>>>(t, Trow);
  pool2_kernel<<<(B_ * 4) / 4, 128, 0, stream>>>(Trow, S);
  init_resid_kernel<<<EG4, EB, 0, stream>>>(t, S, r, p, sc);

  for (int it = 0; it < CG_ITERS; ++it) {
    stencil_kernel<<<SG, 256, 0, stream>>>(p, aff, t);               // t = L p
    pool1_kernel<<<(B_ * 32) / 4, 128, 0, stream>>>(t, Trow);        // WMMA pool
    pool2_kernel<<<(B_ * 4) / 4, 128, 0, stream>>>(Trow, S);         // WMMA pool
    finish_kernel<<<EG4, EB, 0, stream>>>(p, t, S, sc);              // Ap, p.Ap
    scalar1_kernel<<<1, 1, 0, stream>>>(sc);                         // alpha
    axpy_kernel<<<EG4, EB, 0, stream>>>(yout, r, p, t, sc, sc + 4);  // y,r, rr_new
    scalar2_kernel<<<1, 1, 0, stream>>>(sc);                         // beta
    pupdate_kernel<<<EG4, EB, 0, stream>>>(p, r, sc);                // p = r+beta p
  }
}